// CausalAttentionSortNet_1580547971845
// MI455X (gfx1250) — compile-verified
//
#include <hip/hip_runtime.h>
#include <hip/hip_bf16.h>

typedef __attribute__((ext_vector_type(2))) float v2f;
typedef __attribute__((ext_vector_type(8))) float v8f;

#define BH      64
#define NSEQ    4096
#define DIMK    64
#define BUCKETS 64
#define NCOLS   65   // topk(=1) zero-padded column + 64 kv buckets
#define NPAD    80   // 5 N-tiles of 16

__global__ __launch_bounds__(128)
void sortnet_kernel(const float* __restrict__ q,
                    const float* __restrict__ k,
                    float* __restrict__ out) {
  __shared__ float sqs[BUCKETS][DIMK];   // 16 KB: cumavg(q) at bucket starts
  __shared__ float skp[NPAD][DIMK];      // 20 KB: [0]=zero pad, [1..64]=sk, [65..79]=tile pad
  __shared__ float Rs[BUCKETS][NPAD];    // 20 KB: score matrix (scaled)

  const int b   = blockIdx.x;
  const int tid = threadIdx.x;

  // Zero the 16 padding rows of skp (rows {0, 65..79}), 1024 floats / 128 thr.
  #pragma unroll
  for (int z = 0; z < 8; ++z) {
    const int f   = tid * 8 + z;          // 0..1023
    const int ri  = f >> 6;               // 0..15
    const int zr  = (ri == 0) ? 0 : (64 + ri);
    skp[zr][f & 63] = 0.f;
  }

  // ---------------- Phase 1: streaming prefix scans (memory-bound part) ----
  if (tid < 64) {
    // q-scan: dim e, need cumavg at t = i*64 only (t <= 4032)
    const int e = tid;
    const float* qp = q + (size_t)b * NSEQ * DIMK + e;
    float s = 0.f;
    #pragma unroll 1
    for (int t = 0; t <= (BUCKETS - 1) * 64; ++t) {
      __builtin_prefetch(qp + (size_t)(t + 24) * DIMK, 0, 1);
      s += qp[(size_t)t * DIMK];
      if ((t & 63) == 0) sqs[t >> 6][e] = s / (float)(t + 1);
    }
  } else {
    // k-scan: dim e, accumulate cumsum(k)[t]/(t+1) within each bucket
    const int e = tid - 64;
    const float* kp = k + (size_t)b * NSEQ * DIMK + e;
    float s = 0.f, acc = 0.f;
    #pragma unroll 1
    for (int t = 0; t < NSEQ; ++t) {
      __builtin_prefetch(kp + (size_t)(t + 24) * DIMK, 0, 1);
      s += kp[(size_t)t * DIMK];
      acc += s / (float)(t + 1);
      if ((t & 63) == 63) { skp[(t >> 6) + 1][e] = acc; acc = 0.f; }
    }
  }
  __syncthreads();

  // ---------------- Phase 2: R = sq * sk_pad^T * DIM^-0.5 via f32 WMMA -----
  // wave w owns rows [16w, 16w+16); loop over 5 N-tiles; K=64 in steps of 4.
  const int lane    = tid & 31;
  const int wave    = tid >> 5;
  const int rowbase = wave * 16;
  const int mrow    = lane & 15;          // A: M index
  const int khalf   = (lane >> 4) << 1;   // A/B: K sub-offset 0 or 2

  for (int nt = 0; nt < 5; ++nt) {
    const int ncol = nt * 16 + (lane & 15);   // B: N index (padded col j)
    v8f acc = {};
    #pragma unroll
    for (int kb = 0; kb < DIMK; kb += 4) {
      const int k0 = kb + khalf;
      v2f a, bb;
      a.x  = sqs[rowbase + mrow][k0];
      a.y  = sqs[rowbase + mrow][k0 + 1];
      bb.x = skp[ncol][k0];                   // rows 0 and >=65 are zeros
      bb.y = skp[ncol][k0 + 1];
      acc = __builtin_amdgcn_wmma_f32_16x16x4_f32(
          /*neg_a=*/false, a, /*neg_b=*/false, bb,
          /*c_mod=*/(short)0, acc, /*reuse_a=*/false, /*reuse_b=*/false);
    }
    // C/D layout: VGPR r -> M = rowbase + r + 8*(lane>=16), N = nt*16 + lane%16
    #pragma unroll
    for (int r = 0; r < 8; ++r) {
      const int mi = rowbase + r + ((lane >> 4) << 3);
      const int nj = nt * 16 + (lane & 15);
      Rs[mi][nj] = acc[r] * 0.125f;   // * DIM^-0.5
    }
  }
  __syncthreads();

  // ---------------- Phase 3: masked softmax -> one-hot(argmax) * prob ------
  if (tid < 64) {
    const int i = tid;                      // row; valid cols j = 0..i
    float maxv = -3.402823466e38f;
    int   argj = 0;
    for (int j = 0; j <= i; ++j) {
      const float v = Rs[i][j];
      if (v > maxv) { maxv = v; argj = j; }  // strict '>' == first argmax
    }
    float sum = 0.f;
    for (int j = 0; j <= i; ++j)
      sum += expf(Rs[i][j] - maxv);          // masked cols exp underflow to 0
    const float val = 1.0f / sum;            // exp(0)/sum at the argmax
    float* op = out + ((size_t)b * BUCKETS + i) * NCOLS;
    for (int j = 0; j < NCOLS; ++j)
      op[j] = (j == argj) ? val : 0.f;
  }
}

extern "C" void kernel_launch(void* const* d_in, const int* in_sizes, int n_in,
                              void* d_out, int out_size, void* d_ws, size_t ws_size,
                              hipStream_t stream) {
  const float* q = (const float*)d_in[0];
  const float* k = (const float*)d_in[1];
  // d_in[2] = topk (int, ==1) — compile-time constant here
  float* out = (float*)d_out;
  sortnet_kernel<<<BH, 128, 0, stream>>>(q, k, out);
}